// VectorQuantizer_15032385536268
// MI455X (gfx1250) — compile-verified
//
#include <hip/hip_runtime.h>

typedef float v2f __attribute__((ext_vector_type(2)));
typedef float v8f __attribute__((ext_vector_type(8)));

#define K_CODES 512
#define EMB_D   64
#define HW      4096          // 64*64
#define N_PTS   131072        // 32*64*64
#define ROWS_PER_BLOCK 128
#define ESTRIDE 66            // padded LDS row stride for E (bank-conflict avoidance, 8B aligned)

// ---------------------------------------------------------------------------
// Prologue: zero the loss accumulator and precompute ||E_k||^2 into workspace.
// ---------------------------------------------------------------------------
__global__ __launch_bounds__(256) void vq_prep(const float* __restrict__ E,
                                               float* __restrict__ enorm,
                                               float* __restrict__ loss_out) {
  int k = blockIdx.x * blockDim.x + threadIdx.x;
  if (k == 0) loss_out[0] = 0.0f;
  if (k < K_CODES) {
    float s = 0.0f;
#pragma unroll 8
    for (int d = 0; d < EMB_D; ++d) { float e = E[k * EMB_D + d]; s += e * e; }
    enorm[k] = s;
  }
}

// ---------------------------------------------------------------------------
// Fused kernel: distances via fp32 WMMA, argmin, gather E[idx], loss.
// Block = 256 threads (8 wave32), 128 rows per block, grid = N/128 = 1024.
// ---------------------------------------------------------------------------
__global__ __launch_bounds__(256) void vq_main(const float* __restrict__ in,
                                               const float* __restrict__ E,
                                               const float* __restrict__ enorm,
                                               float* __restrict__ out) {
  extern __shared__ float smem[];
  float* Elds   = smem;                       // K_CODES * ESTRIDE floats (padded E)
  float* EnormL = Elds + K_CODES * ESTRIDE;   // K_CODES floats
  float* wPart  = EnormL + K_CODES;           // 8 floats (per-wave loss partials)
  int*   idxL   = (int*)(wPart + 8);          // 128 ints (per-row argmin)

  const int tid = threadIdx.x;
  const int base = blockIdx.x * ROWS_PER_BLOCK;   // global point index of row 0
  const int b    = base >> 12;                    // batch (base / 4096)
  const int hwb  = base & (HW - 1);               // h*W + w of row 0

  // ---- Stage E into LDS (padded row-major) and load code norms ----
  for (int i = tid; i < K_CODES * (EMB_D / 2); i += 256) {
    int k  = i / (EMB_D / 2);
    int d2 = (i % (EMB_D / 2)) * 2;
    float2 v = *(const float2*)&E[k * EMB_D + d2];
    Elds[k * ESTRIDE + d2]     = v.x;
    Elds[k * ESTRIDE + d2 + 1] = v.y;
  }
  for (int i = tid; i < K_CODES; i += 256) EnormL[i] = enorm[i];
  __syncthreads();

  const int wv    = tid >> 5;
  const int lane  = tid & 31;
  const int m     = lane & 15;          // row within the wave's 16-row tile
  const int koff2 = (lane >> 4) * 2;    // K sub-offset per WMMA fp32 A/B layout
  const int hw_row = hwb + wv * 16 + m;

  // ---- Load A fragments once: Ze[row][0..63] in WMMA 16x4 layout ----
  v2f a[16];
#pragma unroll
  for (int s = 0; s < 16; ++s) {
    int d0 = s * 4 + koff2;
    a[s].x = in[(b * EMB_D + d0    ) * HW + hw_row];
    a[s].y = in[(b * EMB_D + d0 + 1) * HW + hw_row];
  }

  // ---- Sweep the 512 codes in 32 groups of 16, tracking per-row best ----
  float bv[8];
  int   bi[8];
#pragma unroll
  for (int r = 0; r < 8; ++r) { bv[r] = 3.4e38f; bi[r] = 0; }

  const int ncol = lane & 15;   // column (code within group) owned by this lane
#pragma unroll 1
  for (int cg = 0; cg < 32; ++cg) {
    const int code = cg * 16 + ncol;
    const float* erow = &Elds[code * ESTRIDE + koff2];
    v8f c = {};
#pragma unroll
    for (int s = 0; s < 16; ++s) {
      v2f bb = *(const v2f*)&erow[s * 4];
      c = __builtin_amdgcn_wmma_f32_16x16x4_f32(
          /*neg_a=*/false, a[s], /*neg_b=*/false, bb,
          /*c_mod=*/(short)0, c, /*reuse_a=*/false, /*reuse_b=*/false);
    }
    const float en = EnormL[code];
#pragma unroll
    for (int r = 0; r < 8; ++r) {
      float dv = en - 2.0f * c[r];          // ||e||^2 - 2 z.e  (||z||^2 dropped)
      if (dv < bv[r]) { bv[r] = dv; bi[r] = code; }
    }
  }

  // ---- Argmin across the 16 lanes sharing each row (C-matrix layout) ----
#pragma unroll
  for (int r = 0; r < 8; ++r) {
    float v  = bv[r];
    int   ix = bi[r];
#pragma unroll
    for (int msk = 8; msk >= 1; msk >>= 1) {
      float ov = __shfl_xor(v, msk, 32);
      int   oi = __shfl_xor(ix, msk, 32);
      if (ov < v || (ov == v && oi < ix)) { v = ov; ix = oi; }  // jnp.argmin: first match
    }
    if ((lane & 15) == 0)            // lane 0 -> row r, lane 16 -> row r+8
      idxL[wv * 16 + (lane >> 4) * 8 + r] = ix;
  }
  __syncthreads();

  // ---- Gather Zq = E[idx] back to [B,D,H,W] + loss accumulation ----
  const int row  = tid & 127;       // local row
  const int dph  = tid >> 7;        // 0 or 1: interleave d across thread halves
  const int code = idxL[row];
  float acc = 0.0f;
  const int gbase = b * EMB_D * HW + hwb + row;
#pragma unroll
  for (int d = dph; d < EMB_D; d += 2) {
    float e = Elds[code * ESTRIDE + d];
    int off = gbase + d * HW;
    float z = in[off];
    out[1 + off] = e;               // out[0] is the loss scalar
    float df = e - z;
    acc += df * df;
  }

  // block reduction of the squared-error partials
#pragma unroll
  for (int msk = 16; msk >= 1; msk >>= 1) acc += __shfl_xor(acc, msk, 32);
  if (lane == 0) wPart[wv] = acc;
  __syncthreads();
  if (tid == 0) {
    float t = 0.0f;
#pragma unroll
    for (int i = 0; i < 8; ++i) t += wPart[i];
    // loss = (1 + beta) * mean((Zq - Ze)^2),  beta = 0.25
    atomicAdd(out, t * (1.25f / 8388608.0f));
  }
}

extern "C" void kernel_launch(void* const* d_in, const int* in_sizes, int n_in,
                              void* d_out, int out_size, void* d_ws, size_t ws_size,
                              hipStream_t stream) {
  const float* in  = (const float*)d_in[0];   // [32, 64, 64, 64] fp32
  const float* E   = (const float*)d_in[1];   // [512, 64] fp32
  float* out   = (float*)d_out;               // [1 + 8388608] fp32
  float* enorm = (float*)d_ws;                // 512 floats of scratch

  const size_t smem_bytes =
      (size_t)(K_CODES * ESTRIDE + K_CODES + 8) * sizeof(float) + 128 * sizeof(int);

  // Host-side attribute set; identical on every call (no static guards).
  hipFuncSetAttribute((const void*)vq_main,
                      hipFuncAttributeMaxDynamicSharedMemorySize,
                      (int)smem_bytes);

  vq_prep<<<2, 256, 0, stream>>>(E, enorm, out);
  vq_main<<<N_PTS / ROWS_PER_BLOCK, 256, smem_bytes, stream>>>(in, E, enorm, out);
}